// PatchSelectiveTransformer_72095321030739
// MI455X (gfx1250) — compile-verified
//
#include <hip/hip_runtime.h>
#include <math.h>

// ---------------------------------------------------------------------------
// PatchSelectiveTransformer on MI455X (gfx1250, wave32, WMMA f32 16x16x4)
//
// Live dataflow (top-k / gather is dead code w.r.t. the output slice):
//   X0[r,l,n,e] -> qkv GEMM -> per-(r,n,h) 32x32 attention -> out proj ->
//   scatter orig -> boundary smoothing.
// ---------------------------------------------------------------------------

typedef __attribute__((ext_vector_type(2))) float v2f;
typedef __attribute__((ext_vector_type(8))) float v8f;

#define QKV_STRIDE 196   // 192 cols + pad (stride%64==4 -> conflict free col walks)
#define X_STRIDE   68    // 64 cols + pad (16B aligned rows for float4 stores)
#define S_STRIDE   33    // 32 cols + pad (odd stride -> conflict free col walks)
#define S_HEAD     (32 * S_STRIDE)          // 1056 floats per head
#define QKV_OFF    0
#define S_OFF      (32 * QKV_STRIDE)        // 6272
#define X_OFF      S_OFF                    // X tile aliases scores region (dead by stage 3)
#define SMEM_FLOATS (S_OFF + 8 * S_HEAD)    // 14720 floats = 58880 bytes

__device__ __forceinline__ v8f wmma4(v2f a, v2f b, v8f c) {
  // V_WMMA_F32_16X16X4_F32 : D = A(16x4,f32) * B(4x16,f32) + C(16x16,f32)
  return __builtin_amdgcn_wmma_f32_16x16x4_f32(
      /*neg_a=*/false, a, /*neg_b=*/false, b,
      /*c_mod=*/(short)0, c, /*reuse_a=*/false, /*reuse_b=*/false);
}

__global__ __launch_bounds__(256)
void fused_attn_kernel(const float* __restrict__ x,        // (2,64,256,256)
                       const float* __restrict__ in_w,     // (192,64)
                       const float* __restrict__ in_b,     // (192)
                       const float* __restrict__ out_w,    // (64,64)
                       const float* __restrict__ out_b,    // (64)
                       float* __restrict__ orig)           // (2,64,256,256) scratch
{
  __shared__ __align__(16) float smem[SMEM_FLOATS];

  const int wg   = blockIdx.x;       // 0..4095
  const int r    = wg >> 8;          // patch row 0..15
  const int n    = wg & 255;         // token index within self slice
  const int c    = n >> 2;           // channel 0..63
  const int s    = n & 3;            // sub-row group 0..3
  const int tid  = threadIdx.x;
  const int wave = tid >> 5;         // 0..7 (== head in attention stages)
  const int lane = tid & 31;
  const int lo   = lane & 15;
  const int hi   = lane >> 4;
  const int kofs = hi << 1;          // K-pair select per WMMA operand layout

  // gentle L2 warm for the weight matrices (global_prefetch_b8)
  __builtin_prefetch(in_w + tid * 48, 0, 0);
  __builtin_prefetch(out_w + tid * 16, 0, 0);

  // ---- stage 1: gather X0 tile (32 rows l, 64 cols e) into LDS -------------
  // X0[l=(b,pw)][e] = x[b, c, r*16 + s*4 + e/16, pw*16 + e%16]
  for (int it = 0; it < 2; ++it) {
    int seg = it * 256 + tid;              // 512 float4 segments
    int l   = seg >> 4;                    // 0..31
    int e4  = (seg & 15) << 2;             // 0,4,...,60
    int b   = l >> 4, pw = l & 15;
    int hrow = r * 16 + s * 4 + (e4 >> 4);
    const float4 v = *(const float4*)&x[(((b << 6) + c) * 256 + hrow) * 256 +
                                        (pw << 4) + (e4 & 15)];
    *(float4*)&smem[X_OFF + l * X_STRIDE + e4] = v;
  }
  __syncthreads();

  // ---- stage 2: qkv = X0 @ W_in^T + b ; q *= dh^-0.5 -----------------------
  // M=32 (2 tiles) x N=192 (12 tiles) x K=64 (16 steps) -> 24 tiles / 8 waves
  for (int tt = wave; tt < 24; tt += 8) {
    int mt = tt / 12, nt = tt % 12;
    int e = (nt << 4) + lo;                          // output channel 0..191
    const float* wr = in_w + e * 64;                 // B[k][n=e] = in_w[e][k]
    const float* xr = &smem[X_OFF + (mt * 16 + lo) * X_STRIDE];
    v8f acc = {};
    for (int k0 = 0; k0 < 64; k0 += 4) {
      v2f a, b;
      a.x = xr[k0 + kofs];      a.y = xr[k0 + kofs + 1];
      b.x = wr[k0 + kofs];      b.y = wr[k0 + kofs + 1];
      acc = wmma4(a, b, acc);
    }
    float bias = in_b[e];
    float scl  = (e < 64) ? 0.35355339059327373f : 1.0f;   // 8^-0.5 on q
#pragma unroll
    for (int j = 0; j < 8; ++j) {
      int row = mt * 16 + hi * 8 + j;                      // l
      smem[QKV_OFF + row * QKV_STRIDE + e] = (acc[j] + bias) * scl;
    }
  }
  __syncthreads();

  // ---- stage 3: scores[h][l][m] = q[l]·k[m]  (head h = wave) ---------------
  {
    int h = wave;
    for (int lt = 0; lt < 2; ++lt)
      for (int mt2 = 0; mt2 < 2; ++mt2) {
        int lrow = lt * 16 + lo;
        int mrow = mt2 * 16 + lo;
        v8f acc = {};
        for (int k0 = 0; k0 < 8; k0 += 4) {
          int d0 = k0 + kofs;
          v2f a, b;
          a.x = smem[QKV_OFF + lrow * QKV_STRIDE + h * 8 + d0];
          a.y = smem[QKV_OFF + lrow * QKV_STRIDE + h * 8 + d0 + 1];
          b.x = smem[QKV_OFF + mrow * QKV_STRIDE + 64 + h * 8 + d0];
          b.y = smem[QKV_OFF + mrow * QKV_STRIDE + 64 + h * 8 + d0 + 1];
          acc = wmma4(a, b, acc);
        }
#pragma unroll
        for (int j = 0; j < 8; ++j) {
          int row = lt * 16 + hi * 8 + j;
          smem[S_OFF + h * S_HEAD + row * S_STRIDE + mt2 * 16 + lo] = acc[j];
        }
      }
  }
  __syncthreads();

  // ---- stage 4: softmax over m (one thread per (h, l) row) -----------------
  {
    int h = tid >> 5, l = tid & 31;
    float* srow = &smem[S_OFF + h * S_HEAD + l * S_STRIDE];
    float mx = -3.4e38f;
    for (int m = 0; m < 32; ++m) mx = fmaxf(mx, srow[m]);
    float sum = 0.f;
    for (int m = 0; m < 32; ++m) { float ev = expf(srow[m] - mx); srow[m] = ev; sum += ev; }
    float inv = 1.f / sum;
    for (int m = 0; m < 32; ++m) srow[m] *= inv;
  }
  __syncthreads();

  // ---- stage 5: o = attn @ v  (head h = wave; o into dead q region) --------
  {
    int h = wave;
    for (int lt = 0; lt < 2; ++lt) {
      v8f acc = {};
      for (int k0 = 0; k0 < 32; k0 += 4) {
        int m0 = k0 + kofs;
        v2f a, b;
        a.x = smem[S_OFF + h * S_HEAD + (lt * 16 + lo) * S_STRIDE + m0];
        a.y = smem[S_OFF + h * S_HEAD + (lt * 16 + lo) * S_STRIDE + m0 + 1];
        float b0 = 0.f, b1 = 0.f;
        if (lo < 8) {                                   // N padded 8 -> 16
          b0 = smem[QKV_OFF + m0 * QKV_STRIDE + 128 + h * 8 + lo];
          b1 = smem[QKV_OFF + (m0 + 1) * QKV_STRIDE + 128 + h * 8 + lo];
        }
        b.x = b0; b.y = b1;
        acc = wmma4(a, b, acc);                         // full EXEC here
      }
#pragma unroll
      for (int j = 0; j < 8; ++j) {
        if (lo < 8) {
          int row = lt * 16 + hi * 8 + j;
          smem[QKV_OFF + row * QKV_STRIDE + h * 8 + lo] = acc[j];
        }
      }
    }
  }
  __syncthreads();

  // ---- stage 6: y = o @ W_out^T + b_out ; scatter to orig ------------------
  {
    int mt = wave >> 2, nt = wave & 3;                  // 8 tiles / 8 waves
    int e = (nt << 4) + lo;                             // output channel 0..63
    const float* wr = out_w + e * 64;
    v8f acc = {};
    for (int k0 = 0; k0 < 64; k0 += 4) {
      v2f a, b;
      a.x = smem[QKV_OFF + (mt * 16 + lo) * QKV_STRIDE + k0 + kofs];
      a.y = smem[QKV_OFF + (mt * 16 + lo) * QKV_STRIDE + k0 + kofs + 1];
      b.x = wr[k0 + kofs];  b.y = wr[k0 + kofs + 1];
      acc = wmma4(a, b, acc);
    }
    float bias = out_b[e];
    int hrow = r * 16 + s * 4 + (e >> 4);
    int wc   = e & 15;
#pragma unroll
    for (int j = 0; j < 8; ++j) {
      int row = mt * 16 + hi * 8 + j;                   // l = b*16 + pw
      int b_ = row >> 4, pw = row & 15;
      orig[(((b_ << 6) + c) * 256 + hrow) * 256 + (pw << 4) + wc] = acc[j] + bias;
    }
  }
}

// ---- boundary smoothing: column rule overrides row rule at intersections ---
__global__ __launch_bounds__(256)
void smooth_kernel(const float* __restrict__ orig, float* __restrict__ out) {
  int i = blockIdx.x * 256 + threadIdx.x;   // 2*64*256*256 = 8388608 elements
  int w = i & 255;
  int h = (i >> 8) & 255;
  float v = orig[i];
  if ((w & 15) == 0 && w != 0)      v = 0.5f * (v + orig[i - 1]);
  else if ((h & 15) == 0 && h != 0) v = 0.5f * (v + orig[i - 256]);
  out[i] = v;
}

extern "C" void kernel_launch(void* const* d_in, const int* in_sizes, int n_in,
                              void* d_out, int out_size, void* d_ws, size_t ws_size,
                              hipStream_t stream) {
  const float* x    = (const float*)d_in[0];
  const float* in_w = (const float*)d_in[1];
  const float* in_b = (const float*)d_in[2];
  const float* ow   = (const float*)d_in[3];
  const float* ob   = (const float*)d_in[4];
  float* out  = (float*)d_out;
  float* orig = (float*)d_ws;   // 33.5 MB scratch for pre-smoothing output

  fused_attn_kernel<<<dim3(16 * 256), dim3(256), 0, stream>>>(x, in_w, in_b, ow, ob, orig);
  smooth_kernel<<<dim3((2 * 64 * 256 * 256) / 256), dim3(256), 0, stream>>>(orig, out);
}